// BarrelShifterRight57WithSticky_76312978916098
// MI455X (gfx1250) — compile-verified
//
#include <hip/hip_runtime.h>
#include <stdint.h>

#define DATA_BITS 57
#define SHIFT_BITS 6
#define ROWS_PER_WAVE 32
#define WAVES_PER_BLOCK 8
#define ROWS_PER_BLOCK (ROWS_PER_WAVE * WAVES_PER_BLOCK)
#define TILE_F (ROWS_PER_WAVE * DATA_BITS)   /* 1824 floats = 7296 bytes, 16B aligned */

// 16-byte GCC vector matching the builtin's parameter type from the diagnostics
typedef int v4i __attribute__((vector_size(16)));

typedef __attribute__((address_space(1))) v4i* g_v4i_p;   // global int4*
typedef __attribute__((address_space(3))) v4i* l_v4i_p;   // LDS int4*
typedef __attribute__((address_space(1))) int* g_i32_p;   // global int*
typedef __attribute__((address_space(3))) int* l_i32_p;   // LDS int*

#if defined(__has_builtin)
#if __has_builtin(__builtin_amdgcn_global_load_async_to_lds_b128) && \
    __has_builtin(__builtin_amdgcn_global_load_async_to_lds_b32)
#define HAVE_ASYNC_LOAD 1
#endif
#if __has_builtin(__builtin_amdgcn_global_store_async_from_lds_b128) && \
    __has_builtin(__builtin_amdgcn_global_store_async_from_lds_b32)
#define HAVE_ASYNC_STORE 1
#endif
#if __has_builtin(__builtin_amdgcn_s_wait_asynccnt)
#define HAVE_WAIT_ASYNC_BUILTIN 1
#endif
#endif
#ifndef HAVE_ASYNC_LOAD
#define HAVE_ASYNC_LOAD 0
#endif
#ifndef HAVE_ASYNC_STORE
#define HAVE_ASYNC_STORE 0
#endif
#ifndef HAVE_WAIT_ASYNC_BUILTIN
#define HAVE_WAIT_ASYNC_BUILTIN 0
#endif

__device__ __forceinline__ void cfence() { asm volatile("" ::: "memory"); }

// generic global VA == 64-bit address; low 32 bits of a generic LDS address == LDS offset
__device__ __forceinline__ g_v4i_p gas_v4(const void* p) { return (g_v4i_p)(uintptr_t)p; }
__device__ __forceinline__ l_v4i_p las_v4(const void* p) { return (l_v4i_p)(uint32_t)(uintptr_t)p; }
__device__ __forceinline__ g_i32_p gas_i32(const void* p) { return (g_i32_p)(uintptr_t)p; }
__device__ __forceinline__ l_i32_p las_i32(const void* p) { return (l_i32_p)(uint32_t)(uintptr_t)p; }

__device__ __forceinline__ void wait_async0() {
#if HAVE_WAIT_ASYNC_BUILTIN
  __builtin_amdgcn_s_wait_asynccnt(0);
  cfence();
#else
  asm volatile("s_wait_asynccnt 0x0" ::: "memory");
#endif
}
__device__ __forceinline__ void wait_ds0() {
  asm volatile("s_wait_dscnt 0x0" ::: "memory");
}

// inputs are exactly 0.0f or 1.0f: exponent bit 23 is set iff value == 1.0f
__device__ __forceinline__ uint32_t fbit(float f) {
  return (__float_as_uint(f) >> 23) & 1u;
}

// pack 57-bit row -> shift -> unpack; returns sticky as float
__device__ __forceinline__ float barrel_row(const float* __restrict__ xin,
                                            const float* __restrict__ srow,
                                            float* __restrict__ xout) {
  uint32_t t = 0;
#pragma unroll
  for (int j = 0; j < SHIFT_BITS; ++j) t |= fbit(srow[j]) << (SHIFT_BITS - 1 - j);

  uint32_t lo = 0, hi = 0;
#pragma unroll
  for (int i = 0; i < 32; ++i) lo |= fbit(xin[i]) << i;
#pragma unroll
  for (int i = 0; i < DATA_BITS - 32; ++i) hi |= fbit(xin[32 + i]) << i;

  uint64_t bits = ((uint64_t)hi << 32) | (uint64_t)lo;
  // out[i] = X[i - t]  ->  low 57 bits of (bits << t); t in [0,63] so shift is defined
  uint64_t sh = (bits << t) & ((1ull << DATA_BITS) - 1ull);
  // dropped bits are X[57-t .. 56]
  uint64_t dropped = (t >= DATA_BITS) ? bits : (bits >> (DATA_BITS - t));
  float sticky = dropped ? 1.0f : 0.0f;

  uint32_t slo = (uint32_t)sh, shi = (uint32_t)(sh >> 32);
#pragma unroll
  for (int i = 0; i < 32; ++i) xout[i] = (float)((slo >> i) & 1u);
#pragma unroll
  for (int i = 0; i < DATA_BITS - 32; ++i) xout[32 + i] = (float)((shi >> i) & 1u);
  return sticky;
}

extern "C" __global__ __launch_bounds__(ROWS_PER_BLOCK)
void barrel57_sticky_kernel(const float* __restrict__ X,
                            const float* __restrict__ S,
                            float* __restrict__ outBits,
                            float* __restrict__ outSticky,
                            int rows) {
  __shared__ __align__(16) float lds[WAVES_PER_BLOCK * TILE_F];  // 58368 B

  const int lane = threadIdx.x & 31;
  const int wave = threadIdx.x >> 5;
  const long long tile = (long long)blockIdx.x * WAVES_PER_BLOCK + wave;
  const long long R0 = tile * ROWS_PER_WAVE;
  if (R0 >= rows) return;                 // idle waves exit; no block barriers used anywhere

  float* lws = &lds[wave * TILE_F];

  if (R0 + ROWS_PER_WAVE <= (long long)rows) {
    // ---------------- fast per-wave tile path (32 rows, 7296 contiguous bytes) -------------
    const float* gx = X + R0 * DATA_BITS;

#if HAVE_ASYNC_LOAD
    // 14 x b128 (512 B per wave-instruction) + 1 x b32 tail (128 B)
#pragma unroll
    for (int i = 0; i < 14; ++i) {
      __builtin_amdgcn_global_load_async_to_lds_b128(
          gas_v4(gx + i * 128 + lane * 4),
          las_v4(lws + i * 128 + lane * 4), 0, 0);
    }
    __builtin_amdgcn_global_load_async_to_lds_b32(
        gas_i32(gx + 1792 + lane),
        las_i32(lws + 1792 + lane), 0, 0);
    wait_async0();          // ASYNCcnt==0: tile resident in LDS
#else
    {
      const float4* g4 = (const float4*)gx;   // 16B aligned: R0 multiple of 32 -> 7296B granules
      float4* l4 = (float4*)lws;
#pragma unroll
      for (int i = 0; i < 14; ++i) l4[i * 32 + lane] = g4[i * 32 + lane];
      lws[1792 + lane] = gx[1792 + lane];
      __builtin_amdgcn_wave_barrier();        // intra-wave: DS pipe is in-order, just stop reordering
      cfence();
    }
#endif

    // each lane owns one row in LDS; stride 57 dwords (odd) -> bank-conflict-free
    const long long row = R0 + lane;
    float* xrow = lws + lane * DATA_BITS;
    float sticky = barrel_row(xrow, S + row * SHIFT_BITS, xrow /* overwrite in place */);
    outSticky[row] = sticky;                  // coalesced b32 store

    float* go = outBits + R0 * DATA_BITS;
#if HAVE_ASYNC_STORE
    wait_ds0();             // DS writes visible to DMA engine before async store reads LDS
#pragma unroll
    for (int i = 0; i < 14; ++i) {
      __builtin_amdgcn_global_store_async_from_lds_b128(
          gas_v4(go + i * 128 + lane * 4),
          las_v4(lws + i * 128 + lane * 4), 0, 0);
    }
    __builtin_amdgcn_global_store_async_from_lds_b32(
        gas_i32(go + 1792 + lane),
        las_i32(lws + 1792 + lane), 0, 0);
    // S_ENDPGM performs an implicit wait-idle, draining ASYNCcnt
#else
    __builtin_amdgcn_wave_barrier();
    cfence();
    {
      const float4* l4 = (const float4*)lws;
      float4* o4 = (float4*)go;
#pragma unroll
      for (int i = 0; i < 14; ++i) o4[i * 32 + lane] = l4[i * 32 + lane];
      go[1792 + lane] = lws[1792 + lane];
    }
#endif
  } else {
    // ---------------- tail path (partial wave tile): direct, guarded, no LDS ----------------
    const long long row = R0 + lane;
    if (row < rows) {
      float tmp[DATA_BITS];
      const float* xg = X + row * DATA_BITS;
#pragma unroll
      for (int i = 0; i < DATA_BITS; ++i) tmp[i] = xg[i];
      float sticky = barrel_row(tmp, S + row * SHIFT_BITS, tmp);
      float* og = outBits + row * DATA_BITS;
#pragma unroll
      for (int i = 0; i < DATA_BITS; ++i) og[i] = tmp[i];
      outSticky[row] = sticky;
    }
  }
}

extern "C" void kernel_launch(void* const* d_in, const int* in_sizes, int n_in,
                              void* d_out, int out_size, void* d_ws, size_t ws_size,
                              hipStream_t stream) {
  const float* X = (const float*)d_in[0];       // (rows, 57) float32 of 0/1
  const float* S = (const float*)d_in[1];       // (rows, 6)  float32 of 0/1
  float* out = (float*)d_out;                   // rows*57 shifted bits, then rows sticky
  const int rows = in_sizes[0] / DATA_BITS;
  float* outSticky = out + (size_t)rows * DATA_BITS;

  const int blocks = (rows + ROWS_PER_BLOCK - 1) / ROWS_PER_BLOCK;
  barrel57_sticky_kernel<<<blocks, ROWS_PER_BLOCK, 0, stream>>>(X, S, out, outSticky, rows);
}